// CrossSignalAnalyzer_71889162601013
// MI455X (gfx1250) — compile-verified
//
#include <hip/hip_runtime.h>
#include <hip/hip_bf16.h>
#include <math.h>

typedef __attribute__((ext_vector_type(2))) float v2f;
typedef __attribute__((ext_vector_type(8))) float v8f;

#define NPART 4096
#define TPB   256
#define JSPLIT 8

// ---------------------------------------------------------------------------
// Kernel 1: per-particle derived quantities.
// der layout per particle (8 floats = 2x float4): [t, ra, dec, psi, mc, f_isco, dist, 0]
// ---------------------------------------------------------------------------
__global__ __launch_bounds__(TPB) void derive_kernel(const float* __restrict__ p,
                                                     float* __restrict__ der, int n) {
    int i = blockIdx.x * blockDim.x + threadIdx.x;
    if (i >= n) return;
    const float* q = p + i * 15;
    float m1 = q[0] * 95.0f + 5.0f;
    float m2 = q[1] * 95.0f + 5.0f;
    float mc = powf(m1 * m2, 0.6f) / powf(m1 + m2, 0.2f);
    float fisco = 220.0f / (m1 + m2);
    float dist = q[2] * 2950.0f + 50.0f;
    float* d = der + i * 8;
    d[0] = q[5];   // t
    d[1] = q[3];   // ra
    d[2] = q[4];   // dec
    d[3] = q[7];   // psi
    d[4] = mc;
    d[5] = fisco;
    d[6] = dist;
    d[7] = 0.0f;
}

// ---------------------------------------------------------------------------
// Kernel 2: pairwise feature sums. grid = (n/TPB row-blocks, JSPLIT j-chunks).
// Each thread owns row i, walks its j-chunk in LDS tiles of TPB particles.
// Writes partial[js][i][8]. No atomics -> bitwise deterministic.
// ---------------------------------------------------------------------------
__global__ __launch_bounds__(TPB) void pairwise_kernel(const float* __restrict__ der,
                                                       float* __restrict__ partial, int n) {
    __shared__ float4 tile[TPB][2];
    const int i  = blockIdx.x * TPB + threadIdx.x;
    const int js = blockIdx.y;
    const int jchunk = n / JSPLIT;

    const float4* d4 = (const float4*)der;
    float4 me0 = d4[i * 2 + 0];   // t, ra, dec, psi
    float4 me1 = d4[i * 2 + 1];   // mc, fisco, dist, -
    const float ti = me0.x, ra = me0.y, dec = me0.z, psi = me0.w;
    const float mc = me1.x, fi = me1.y, di = me1.z;

    float acc0 = 0.f, acc1 = 0.f, acc2 = 0.f, acc3 = 0.f;
    float acc4 = 0.f, acc5 = 0.f, acc6 = 0.f, acc7 = 0.f;

    for (int jb = js * jchunk; jb < (js + 1) * jchunk; jb += TPB) {
        tile[threadIdx.x][0] = d4[(jb + threadIdx.x) * 2 + 0];
        tile[threadIdx.x][1] = d4[(jb + threadIdx.x) * 2 + 1];
        __syncthreads();
        #pragma unroll 4
        for (int jj = 0; jj < TPB; ++jj) {
            float4 o0 = tile[jj][0];
            float4 o1 = tile[jj][1];
            float dt   = fabsf(ti  - o0.x);
            float dra  = fabsf(ra  - o0.y);
            float ddec = fabsf(dec - o0.z);
            float dpsi = fabsf(psi - o0.w);
            float sky  = sqrtf(dra * dra + ddec * ddec);
            float msim = 1.0f / (1.0f + fabsf(mc - o1.x) * (1.0f / 30.0f));
            float fov  = expf(-fabsf(fi - o1.y) * 0.01f);
            float dj   = o1.z;
            float drat = fminf(di, dj) / fmaxf(di, dj);
            acc0 += dt;   acc1 += sky;  acc2 += msim; acc3 += fov;
            acc4 += drat; acc5 += dpsi; acc6 += dra;  acc7 += ddec;
        }
        __syncthreads();
    }
    float* o = partial + ((size_t)js * n + i) * 8;
    o[0] = acc0; o[1] = acc1; o[2] = acc2; o[3] = acc3;
    o[4] = acc4; o[5] = acc5; o[6] = acc6; o[7] = acc7;
}

// ---------------------------------------------------------------------------
// Kernel 3: reduce JSPLIT partials, subtract diag, scale by 1/(n-1) -> x[n][8]
// ---------------------------------------------------------------------------
__global__ __launch_bounds__(TPB) void reduce_kernel(const float* __restrict__ partial,
                                                     float* __restrict__ x, int n) {
    int idx = blockIdx.x * blockDim.x + threadIdx.x;   // over n*8
    if (idx >= n * 8) return;
    int f = idx & 7;
    float s = 0.0f;
    #pragma unroll
    for (int js = 0; js < JSPLIT; ++js) s += partial[(size_t)js * n * 8 + idx];
    // diag = [0,0,1,1,1,0,0,0] (self-pair contributions of mass_sim/freq_ov/dist_ratio)
    float diag = (f == 2 || f == 3 || f == 4) ? 1.0f : 0.0f;
    x[idx] = (s - diag) * (1.0f / (float)(n - 1));
}

// ---------------------------------------------------------------------------
// Kernel 4: MLP with f32 WMMA (V_WMMA_F32_16X16X4_F32), one wave32 per 16 rows.
//   h = LN(x@W1 + b1) ; gelu(exact) ; out = h@W2 + b2
// A layout 16x4 f32: lanes 0-15 rows M, vgpr0=(K0|K2 per half), vgpr1=(K1|K3).
// C/D 16x16 f32: vgpr r -> row r (lanes 0-15) / row r+8 (lanes 16-31).
// ---------------------------------------------------------------------------
__device__ __forceinline__ v8f wmma4(v2f a, v2f b, v8f c) {
    return __builtin_amdgcn_wmma_f32_16x16x4_f32(false, a, false, b, (short)0, c,
                                                 false, false);
}

__global__ __launch_bounds__(32) void mlp_kernel(const float* __restrict__ x,
                                                 const float* __restrict__ W1,
                                                 const float* __restrict__ b1,
                                                 const float* __restrict__ lng,
                                                 const float* __restrict__ lnb,
                                                 const float* __restrict__ W2,
                                                 const float* __restrict__ b2,
                                                 float* __restrict__ out) {
    const int lane = threadIdx.x;
    const int half = lane >> 4;        // 0: lanes 0-15, 1: lanes 16-31
    const int l16  = lane & 15;
    const int row0 = blockIdx.x * 16;

    // ---- A tile for layer 1 (x is 16x8 per block) ----
    const float* xr = x + (size_t)(row0 + l16) * 8;
    v2f a0, a1;
    a0.x = xr[half * 2 + 0];     a0.y = xr[half * 2 + 1];      // K 0..3
    a1.x = xr[4 + half * 2 + 0]; a1.y = xr[4 + half * 2 + 1];  // K 4..7

    // ---- B tiles from W1 (8x32 row-major): k-chunk c, n-tile t ----
    const int c0 = l16, c1 = 16 + l16;
    const int kr = half * 2;
    v2f b00, b01, b10, b11;
    b00.x = W1[(kr + 0) * 32 + c0]; b00.y = W1[(kr + 1) * 32 + c0];
    b01.x = W1[(kr + 0) * 32 + c1]; b01.y = W1[(kr + 1) * 32 + c1];
    b10.x = W1[(kr + 4) * 32 + c0]; b10.y = W1[(kr + 5) * 32 + c0];
    b11.x = W1[(kr + 4) * 32 + c1]; b11.y = W1[(kr + 5) * 32 + c1];

    v8f d0 = {}, d1 = {};
    d0 = wmma4(a0, b00, d0);
    d0 = wmma4(a1, b10, d0);
    d1 = wmma4(a0, b01, d1);
    d1 = wmma4(a1, b11, d1);

    // ---- bias + LayerNorm (rows live across the 16 lanes of each half) ----
    const float bia0 = b1[c0], bia1 = b1[c1];
    const float g0 = lng[c0], g1 = lng[c1], e0 = lnb[c0], e1 = lnb[c1];

    __shared__ float hl[16][33];   // padded h tile for layer-2 A re-layout

    #pragma unroll
    for (int r = 0; r < 8; ++r) {
        float u0 = d0[r] + bia0;
        float u1 = d1[r] + bia1;
        float s  = u0 + u1;
        float sq = u0 * u0 + u1 * u1;
        #pragma unroll
        for (int m = 1; m < 16; m <<= 1) {     // 16-lane reduce (stays in half)
            s  += __shfl_xor(s,  m, 32);
            sq += __shfl_xor(sq, m, 32);
        }
        float mean = s * (1.0f / 32.0f);
        float var  = sq * (1.0f / 32.0f) - mean * mean;
        float rstd = rsqrtf(var + 1e-5f);
        u0 = (u0 - mean) * rstd * g0 + e0;
        u1 = (u1 - mean) * rstd * g1 + e1;
        // exact gelu: u * 0.5 * (1 + erf(u/sqrt(2)))
        u0 = u0 * 0.5f * (1.0f + erff(u0 * 0.70710678118654752f));
        u1 = u1 * 0.5f * (1.0f + erff(u1 * 0.70710678118654752f));
        hl[r + 8 * half][c0] = u0;
        hl[r + 8 * half][c1] = u1;
    }
    __syncthreads();

    // ---- layer 2: out(16x16) = h(16x32) @ W2(32x16) + b2, K = 8 chunks of 4 ----
    v8f o = {};
    #pragma unroll
    for (int c = 0; c < 8; ++c) {
        int k = c * 4 + half * 2;
        v2f a; a.x = hl[l16][k + 0]; a.y = hl[l16][k + 1];
        v2f b; b.x = W2[(k + 0) * 16 + l16]; b.y = W2[(k + 1) * 16 + l16];
        o = wmma4(a, b, o);
    }
    const float bo = b2[l16];
    #pragma unroll
    for (int r = 0; r < 8; ++r) {
        out[(size_t)(row0 + r + 8 * half) * 16 + l16] = o[r] + bo;
    }
}

// ---------------------------------------------------------------------------
extern "C" void kernel_launch(void* const* d_in, const int* in_sizes, int n_in,
                              void* d_out, int out_size, void* d_ws, size_t ws_size,
                              hipStream_t stream) {
    const float* p   = (const float*)d_in[0];
    const float* W1  = (const float*)d_in[1];
    const float* b1  = (const float*)d_in[2];
    const float* lng = (const float*)d_in[3];
    const float* lnb = (const float*)d_in[4];
    const float* W2  = (const float*)d_in[5];
    const float* b2  = (const float*)d_in[6];
    float* out = (float*)d_out;

    const int n = in_sizes[0] / 15;   // 4096

    float* der     = (float*)d_ws;                       // n*8
    float* partial = der + (size_t)n * 8;                // JSPLIT*n*8
    float* x       = partial + (size_t)JSPLIT * n * 8;   // n*8

    derive_kernel<<<(n + TPB - 1) / TPB, TPB, 0, stream>>>(p, der, n);

    dim3 pg(n / TPB, JSPLIT);
    pairwise_kernel<<<pg, TPB, 0, stream>>>(der, partial, n);

    reduce_kernel<<<(n * 8 + TPB - 1) / TPB, TPB, 0, stream>>>(partial, x, n);

    mlp_kernel<<<n / 16, 32, 0, stream>>>(x, W1, b1, lng, lnb, W2, b2, out);
}